// DynamicSpatialEncoder_13941463842838
// MI455X (gfx1250) — compile-verified
//
#include <hip/hip_runtime.h>

typedef float v2f __attribute__((ext_vector_type(2)));
typedef float v8f __attribute__((ext_vector_type(8)));

#define DCH   128
#define BATCH 16384
#define BN_EPS 1e-5f

// One wave computes Out(16x128) = relu?( A(16x128) @ W(128x128) + bias )
// using V_WMMA_F32_16X16X4_F32 tiles (exact f32 math).
// A layout per ISA: lane%16 = M row; half-wave picks K pair {0,1} vs {2,3}.
// B layout: VGPR j, half h -> K = j + 2h; lane%16 = N column.
// C/D layout: VGPR j, half h -> M = j + 8h; lane%16 = N.
__device__ __forceinline__ void wave_gemm_16x128(
    const float* __restrict__ lA,    // LDS, 16x128 row-major
    const float* __restrict__ W,     // global, 128x128 row-major (k-major like numpy)
    const float* __restrict__ bias,  // global, 128
    bool do_relu,
    float* __restrict__ lOut)        // LDS, 16x128 row-major
{
  const int lane = (int)threadIdx.x;   // 0..31, wave32
  const int half = lane >> 4;          // 0 or 1
  const int l16  = lane & 15;
  const int m    = l16;                // A row handled by this lane

  for (int nt = 0; nt < 8; ++nt) {
    const int n = nt * 16 + l16;       // output column for this lane
    const float bval = bias[n];        // same bias for every row of the tile
    v8f acc;
#pragma unroll
    for (int j = 0; j < 8; ++j) acc[j] = bval;

#pragma unroll 4
    for (int ks = 0; ks < 32; ++ks) {
      const int kb = ks * 4 + half * 2;  // this half-wave's K pair
      v2f a, b;
      a.x = lA[m * DCH + kb];
      a.y = lA[m * DCH + kb + 1];
      b.x = W[kb * DCH + n];
      b.y = W[(kb + 1) * DCH + n];
      acc = __builtin_amdgcn_wmma_f32_16x16x4_f32(
          /*neg_a=*/false, a, /*neg_b=*/false, b,
          /*c_mod=*/(short)0, acc, /*reuse_a=*/false, /*reuse_b=*/false);
    }

#pragma unroll
    for (int j = 0; j < 8; ++j) {
      float v = acc[j];
      if (do_relu) v = fmaxf(v, 0.0f);
      lOut[(j + 8 * half) * DCH + n] = v;
    }
  }
}

// Batch-norm over the (B,21,D) tensor reduces to exact weighted stats of the
// two distinct rows: root (weight 1/21) and neighbor (weight 20/21).
__device__ __forceinline__ void bn_two_rows(
    const float* __restrict__ lH, const float* __restrict__ g,
    const float* __restrict__ be, int c, float* xr, float* xn)
{
  const float hr = lH[c];            // row 0 (root)
  const float hn = lH[DCH + c];      // row 1 (neighbor)
  const float mean = (hr + 20.0f * hn) * (1.0f / 21.0f);
  const float var  = (hr * hr + 20.0f * hn * hn) * (1.0f / 21.0f) - mean * mean;
  const float inv  = rsqrtf(var + BN_EPS);
  *xr = (hr - mean) * inv * g[c] + be[c];
  *xn = (hn - mean) * inv * g[c] + be[c];
}

__global__ __launch_bounds__(32)
void gin_compute_kernel(const float* __restrict__ Wp,   const float* __restrict__ bp,
                        const float* __restrict__ W1_0, const float* __restrict__ b1_0,
                        const float* __restrict__ W2_0, const float* __restrict__ b2_0,
                        const float* __restrict__ g0,   const float* __restrict__ be0,
                        const float* __restrict__ W1_1, const float* __restrict__ b1_1,
                        const float* __restrict__ W2_1, const float* __restrict__ b2_1,
                        const float* __restrict__ g1,   const float* __restrict__ be1,
                        float* __restrict__ out_vec)
{
  __shared__ float lA[16 * DCH];   // MLP input tile (rows 2..15 stay zero)
  __shared__ float lT[16 * DCH];   // post-relu intermediate
  __shared__ float lH[16 * DCH];   // post-MLP

  const int lane = (int)threadIdx.x;

  // zero pad rows 2..15 of the A tile (only once; GEMMs never write lA)
  for (int i = lane; i < 14 * DCH; i += 32) lA[2 * DCH + i] = 0.0f;

  // layer-0 pre-MLP rows: root = 21*feat, neighbor = 2*feat, feat = Wp + bp
  for (int c = lane; c < DCH; c += 32) {
    const float f = Wp[c] + bp[c];
    lA[c]       = 21.0f * f;
    lA[DCH + c] = 2.0f * f;
  }
  __syncthreads();

  // ---- GIN layer 0 ----
  wave_gemm_16x128(lA, W1_0, b1_0, /*relu=*/true,  lT);
  __syncthreads();
  wave_gemm_16x128(lT, W2_0, b2_0, /*relu=*/false, lH);
  __syncthreads();

  for (int c = lane; c < DCH; c += 32) {
    float xr, xn;
    bn_two_rows(lH, g0, be0, c, &xr, &xn);
    lA[c]       = xr + 20.0f * xn;   // layer-1 root pre-MLP row
    lA[DCH + c] = xn + xr;           // layer-1 neighbor pre-MLP row
  }
  __syncthreads();

  // ---- GIN layer 1 ----
  wave_gemm_16x128(lA, W1_1, b1_1, /*relu=*/true,  lT);
  __syncthreads();
  wave_gemm_16x128(lT, W2_1, b2_1, /*relu=*/false, lH);
  __syncthreads();

  for (int c = lane; c < DCH; c += 32) {
    float xr, xn;
    bn_two_rows(lH, g1, be1, c, &xr, &xn);
    out_vec[c] = xr;                 // final output = normalized root row
  }
}

// Bandwidth-bound part: broadcast the 128-float vector to all 16384 rows.
// 8 MB of stores -> ~0.36 us at 23.3 TB/s; the 512B vector stays L2-hot.
__global__ __launch_bounds__(256)
void broadcast_kernel(const float4* __restrict__ vec4, float4* __restrict__ out4)
{
  const int idx = (int)(blockIdx.x * blockDim.x + threadIdx.x); // float4 index
  out4[idx] = vec4[idx & 31];       // 128 floats = 32 float4s per row
}

extern "C" void kernel_launch(void* const* d_in, const int* in_sizes, int n_in,
                              void* d_out, int out_size, void* d_ws, size_t ws_size,
                              hipStream_t stream)
{
  (void)in_sizes; (void)n_in; (void)out_size; (void)ws_size;
  // setup_inputs order: 0 node_ids, 1 times, 2 neighbors (all unused by math),
  // 3 Wp, 4 bp, 5 W1_0, 6 b1_0, 7 W2_0, 8 b2_0, 9 g0, 10 be0,
  // 11 W1_1, 12 b1_1, 13 W2_1, 14 b2_1, 15 g1, 16 be1
  const float* Wp   = (const float*)d_in[3];
  const float* bp   = (const float*)d_in[4];
  const float* W1_0 = (const float*)d_in[5];
  const float* b1_0 = (const float*)d_in[6];
  const float* W2_0 = (const float*)d_in[7];
  const float* b2_0 = (const float*)d_in[8];
  const float* g0   = (const float*)d_in[9];
  const float* be0  = (const float*)d_in[10];
  const float* W1_1 = (const float*)d_in[11];
  const float* b1_1 = (const float*)d_in[12];
  const float* W2_1 = (const float*)d_in[13];
  const float* b2_1 = (const float*)d_in[14];
  const float* g1   = (const float*)d_in[15];
  const float* be1  = (const float*)d_in[16];

  float* vec = (float*)d_ws;        // 128 floats of scratch
  float* out = (float*)d_out;       // 16384 x 128 f32

  hipLaunchKernelGGL(gin_compute_kernel, dim3(1), dim3(32), 0, stream,
                     Wp, bp, W1_0, b1_0, W2_0, b2_0, g0, be0,
                     W1_1, b1_1, W2_1, b2_1, g1, be1, vec);

  const int n4 = BATCH * DCH / 4;   // 524288 float4 stores
  hipLaunchKernelGGL(broadcast_kernel, dim3(n4 / 256), dim3(256), 0, stream,
                     (const float4*)vec, (float4*)out);
}